// BilinearInteractionLayer_66872640798983
// MI455X (gfx1250) — compile-verified
//
#include <hip/hip_runtime.h>

typedef __attribute__((ext_vector_type(2))) float v2f;
typedef __attribute__((ext_vector_type(4))) float v4f;
typedef __attribute__((ext_vector_type(8))) float v8f;

#define NUM_FIELDS 40
#define EMBED      64
#define NUM_PAIRS  780          // 40*39/2
#define BATCH      2048
#define TILE_B     128          // batch rows per workgroup
#define LDS_STRIDE 68           // 64 floats + 4 pad -> bank-conflict-free frags

__global__ __launch_bounds__(256)
void bilinear_wmma_f32(const float* __restrict__ fe,   // [2048,40,64]
                       const float* __restrict__ W,    // [780,64,64]
                       float* __restrict__ out)        // [2048,780,64]
{
    __shared__ float ldsVi[TILE_B * LDS_STRIDE];   // 34816 B
    __shared__ float ldsW [EMBED  * LDS_STRIDE];   // 17408 B

    const int bid       = blockIdx.x;
    const int p         = bid >> 4;       // pair index: 16 consecutive blocks share W[p]
    const int tb        = bid & 15;       // batch tile index
    const int batchBase = tb * TILE_B;

    // Decode upper-triangular (i, j), k=1, from flat pair index p (uniform -> SALU loop)
    int i = 0, rem = p;
    while (rem >= (NUM_FIELDS - 1 - i)) { rem -= (NUM_FIELDS - 1 - i); ++i; }
    const int j = i + 1 + rem;

    const int t   = threadIdx.x;
    const int seg = t & 15;               // float4 segment within a 64-float row
    const int rw  = t >> 4;               // 0..15

    // ---- Stage Vi tile (TILE_B x 64, strided gather in global) into LDS, b128 coalesced
    #pragma unroll
    for (int r = rw; r < TILE_B; r += 16) {
        const float* src = fe + ((size_t)(batchBase + r) * NUM_FIELDS + i) * EMBED + seg * 4;
        *(v4f*)&ldsVi[r * LDS_STRIDE + seg * 4] = *(const v4f*)src;
    }
    // ---- Stage W[p] (64 x 64) into LDS
    #pragma unroll
    for (int f = rw; f < EMBED; f += 16) {
        const float* src = W + (size_t)p * (EMBED * EMBED) + f * EMBED + seg * 4;
        *(v4f*)&ldsW[f * LDS_STRIDE + seg * 4] = *(const v4f*)src;
    }
    __syncthreads();

    // ---- WMMA: each wave does a 16(batch) x 64(feature) tile, K = 64
    const int wave    = t >> 5;           // 0..7 (wave32)
    const int lane    = t & 31;
    const int l15     = lane & 15;
    const int hi      = lane >> 4;        // 0 or 1
    const int rowBase = wave * 16;

    v8f acc0 = {}, acc1 = {}, acc2 = {}, acc3 = {};

    const float* aRow = &ldsVi[(rowBase + l15) * LDS_STRIDE];
    const float* b0p  = &ldsW[( 0 + l15) * LDS_STRIDE];
    const float* b1p  = &ldsW[(16 + l15) * LDS_STRIDE];
    const float* b2p  = &ldsW[(32 + l15) * LDS_STRIDE];
    const float* b3p  = &ldsW[(48 + l15) * LDS_STRIDE];

    #pragma unroll 4
    for (int k = 0; k < EMBED; k += 4) {
        const int kc = k + hi * 2;        // A: lanes 0-15 hold K={k,k+1}, 16-31 hold K={k+2,k+3}
        v2f a  = *(const v2f*)&aRow[kc];
        v2f b0 = *(const v2f*)&b0p[kc];   // B[k][n] = W[p][n][k]: same striping, N = l15
        v2f b1 = *(const v2f*)&b1p[kc];
        v2f b2 = *(const v2f*)&b2p[kc];
        v2f b3 = *(const v2f*)&b3p[kc];
        acc0 = __builtin_amdgcn_wmma_f32_16x16x4_f32(false, a, false, b0, (short)0, acc0, false, false);
        acc1 = __builtin_amdgcn_wmma_f32_16x16x4_f32(false, a, false, b1, (short)0, acc1, false, false);
        acc2 = __builtin_amdgcn_wmma_f32_16x16x4_f32(false, a, false, b2, (short)0, acc2, false, false);
        acc3 = __builtin_amdgcn_wmma_f32_16x16x4_f32(false, a, false, b3, (short)0, acc3, false, false);
    }

    // ---- Epilogue: scale by Vj and store. D layout: VGPR v -> row v + 8*hi, col l15.
    #pragma unroll
    for (int v = 0; v < 8; ++v) {
        const int    b      = batchBase + rowBase + v + 8 * hi;
        const float* vjRow  = fe  + ((size_t)b * NUM_FIELDS + j) * EMBED;
        float*       outRow = out + ((size_t)b * NUM_PAIRS  + p) * EMBED;
        outRow[ 0 + l15] = acc0[v] * vjRow[ 0 + l15];
        outRow[16 + l15] = acc1[v] * vjRow[16 + l15];
        outRow[32 + l15] = acc2[v] * vjRow[32 + l15];
        outRow[48 + l15] = acc3[v] * vjRow[48 + l15];
    }
}

extern "C" void kernel_launch(void* const* d_in, const int* in_sizes, int n_in,
                              void* d_out, int out_size, void* d_ws, size_t ws_size,
                              hipStream_t stream) {
    (void)in_sizes; (void)n_in; (void)out_size; (void)d_ws; (void)ws_size;
    const float* fe  = (const float*)d_in[0];   // feature_emb [2048,40,64] fp32
    const float* W   = (const float*)d_in[1];   // W [780,64,64] fp32
    float*       out = (float*)d_out;           // [2048,780,64] fp32

    dim3 grid(NUM_PAIRS * (BATCH / TILE_B));    // 780 * 16 = 12480 blocks
    dim3 block(256);                            // 8 wave32 waves
    bilinear_wmma_f32<<<grid, block, 0, stream>>>(fe, W, out);
}